// KMeans_9414568313245
// MI455X (gfx1250) — compile-verified
//
#include <hip/hip_runtime.h>

typedef __attribute__((ext_vector_type(2))) float v2f;
typedef __attribute__((ext_vector_type(8))) float v8f;

constexpr int   kNS   = 1000000;
constexpr int   kNF   = 64;
constexpr int   kNK   = 64;
constexpr int   kIter = 30;
constexpr float kTol  = 1e-5f;

constexpr int kTR = 64;          // tile rows per block (4 waves x 16 rows)
constexpr int kXP = 68;          // x-tile LDS pitch (floats) — avoids stride-64 bank conflicts
constexpr int kCP = 68;          // centers LDS pitch
constexpr int kPSZ = kNK * kNF + kNK + 1;   // per-block partial: sums + counts + pot = 4161

// workspace layout (float offsets)
constexpr int kWC   = 0;                    // centers [64*64]
constexpr int kWCN  = kNK * kNF;            // 4096: center norms [64]
constexpr int kWPP  = kWCN + kNK;           // 4160: prev potential
constexpr int kWST  = kWPP + 1;             // 4161: stopped flag
constexpr int kWXN  = 4352;                 // x row norms [kNS]
constexpr int kWPART = kWXN + kNS;          // per-block partials

constexpr int kTiles = kNS / kTR;           // 15625 exact

__device__ __forceinline__ unsigned lds_off(const void* p) {
    // generic pointer to LDS: aperture in high 32 bits, LDS byte offset in low 32
    return (unsigned)(unsigned long long)(uintptr_t)p;
}

// ---------------------------------------------------------------- init state
__global__ void km_init_kernel(const float* __restrict__ minit, float* __restrict__ ws) {
    int g = blockIdx.x * blockDim.x + threadIdx.x;
    if (g < kNK * kNF) ws[kWC + g] = minit[g];
    if (g < kNK) {
        float s = 0.0f;
        #pragma unroll 8
        for (int f = 0; f < kNF; ++f) { float v = minit[g * kNF + f]; s += v * v; }
        ws[kWCN + g] = s;
    }
    if (g == 0) { ws[kWPP] = __builtin_inff(); ws[kWST] = 0.0f; }
}

// ---------------------------------------------------------- x row norms (x·x)
__global__ void km_xn_kernel(const float* __restrict__ x, float* __restrict__ ws) {
    int wave = threadIdx.x >> 5, lane = threadIdx.x & 31;
    int row = blockIdx.x * 8 + wave;
    if (row >= kNS) return;
    float2 v = *(const float2*)(x + row * kNF + lane * 2);
    float s = v.x * v.x + v.y * v.y;
    for (int o = 16; o >= 1; o >>= 1) s += __shfl_down(s, o);
    if (lane == 0) ws[kWXN + row] = s;
}

// -------------------------------------------- assignment + partial reduction
__global__ __launch_bounds__(128)
void km_assign_kernel(const float* __restrict__ x, int* __restrict__ ixs,
                      float* __restrict__ ws) {
    __shared__ float xtile[2][kTR * kXP];   // 2 x 17408 B, double buffered (async DMA)
    __shared__ float xnl[2][kTR];
    __shared__ float clds[kNK * kCP];       // 17408 B
    __shared__ float csums[kNK * kNF];      // 16384 B
    __shared__ float ccnt[kNK];
    __shared__ float cnl[kNK];
    __shared__ float bestd[kTR];
    __shared__ int   bestk[kTR];

    if (ws[kWST] != 0.0f) return;           // frozen after convergence

    const int tid = threadIdx.x;
    const int lane = tid & 31;
    // stage centers + zero accumulators (once per block)
    for (int e = tid; e < kNK * kNF; e += 128) {
        clds[(e >> 6) * kCP + (e & 63)] = ws[kWC + e];
        csums[e] = 0.0f;
    }
    if (tid < kNK) { cnl[tid] = ws[kWCN + tid]; ccnt[tid] = 0.0f; }

    // issue one tile's async global->LDS DMA: 8 x b128 (x tile) + 1 x b64 (xn strip)
    // => exactly 9 ASYNCcnt increments per wave per tile (loads retire in order)
    auto issue_tile = [&](int tile, int b) {
        const int rbase = tile * kTR;
        #pragma unroll
        for (int i = 0; i < 8; ++i) {
            int g = tid + 128 * i;
            int r = g >> 4, c4 = (g & 15) * 4;
            unsigned l = lds_off(&xtile[b][r * kXP + c4]);
            unsigned long long ga =
                (unsigned long long)(uintptr_t)(x + (rbase + r) * kNF + c4);
            asm volatile("global_load_async_to_lds_b128 %0, %1, off"
                         :: "v"(l), "v"(ga) : "memory");
        }
        {   // xn strip: 32 lanes x 8B (all waves issue it -> uniform count; same data)
            unsigned l = lds_off(&xnl[b][lane * 2]);
            unsigned long long ga =
                (unsigned long long)(uintptr_t)(ws + kWXN + rbase + lane * 2);
            asm volatile("global_load_async_to_lds_b64 %0, %1, off"
                         :: "v"(l), "v"(ga) : "memory");
        }
    };

    const int hi = lane >> 4, n0 = lane & 15;
    const int wr0 = (tid >> 5) * 16;        // wave's 16-row strip inside tile
    float potacc = 0.0f;

    issue_tile(blockIdx.x, 0);              // prologue: fill buffer 0
    int buf = 0;
    for (int tile = blockIdx.x; tile < kTiles; tile += gridDim.x) {
        const int rbase = tile * kTR;
        const int next = tile + (int)gridDim.x;
        const bool hasNext = next < kTiles;
        if (hasNext) issue_tile(next, buf ^ 1);          // overlap DMA with compute
        // L2 warm-up two tiles ahead (global_prefetch_b8)
        int pf = next + (int)gridDim.x;
        if (pf < kTiles) __builtin_prefetch(x + pf * kTR * kNF + tid * 32, 0, 1);
        // wait for current buffer only: oldest 9 async ops (in-order retirement)
        if (hasNext) asm volatile("s_wait_asynccnt 0x9" ::: "memory");
        else         asm volatile("s_wait_asynccnt 0x0" ::: "memory");
        __syncthreads();

        // G = Xtile · Cᵀ with fp32 WMMA 16x16x4, 16 k-steps, 4 cluster tiles
        v8f acc[4] = {};
        #pragma unroll
        for (int k = 0; k < 16; ++k) {
            v2f a = *(const v2f*)&xtile[buf][(wr0 + n0) * kXP + 4 * k + 2 * hi];
            #pragma unroll
            for (int t = 0; t < 4; ++t) {
                v2f b = *(const v2f*)&clds[(n0 + 16 * t) * kCP + 4 * k + 2 * hi];
                acc[t] = __builtin_amdgcn_wmma_f32_16x16x4_f32(
                    false, a, false, b, (short)0, acc[t], false, false);
            }
        }

        // per-row argmin over 64 clusters (D layout: vgpr j -> M=j+8*hi, N=n0)
        #pragma unroll
        for (int j = 0; j < 8; ++j) {
            const int rloc = wr0 + j + 8 * hi;
            const float xnv = xnl[buf][rloc];
            float bd = 3.4e38f; int bk = 0;
            #pragma unroll
            for (int t = 0; t < 4; ++t) {
                int c = n0 + 16 * t;
                float d = fmaxf(xnv + cnl[c] - 2.0f * acc[t][j], 0.0f);
                if (d < bd) { bd = d; bk = c; }       // ascending c -> first-index ties
            }
            for (int s = 8; s >= 1; s >>= 1) {        // butterfly within 16-lane half
                float od = __shfl_xor(bd, s);
                int   ok = __shfl_xor(bk, s);
                if (od < bd || (od == bd && ok < bk)) { bd = od; bk = ok; }
            }
            if (n0 == 0) {
                bestk[rloc] = bk; bestd[rloc] = bd;
                atomicAdd(&ccnt[bk], 1.0f);
            }
        }
        __syncthreads();

        // scatter x rows into per-cluster sums (ds_add_f32), write indices
        const int f = tid & 63, rg = tid >> 6;
        for (int r = rg; r < kTR; r += 2)
            atomicAdd(&csums[bestk[r] * kNF + f], xtile[buf][r * kXP + f]);
        if (tid < kTR) ixs[rbase + tid] = bestk[tid];
        if (tid == 0) {                                // deterministic pot order
            for (int r = 0; r < kTR; ++r) potacc += bestd[r];
        }
        __syncthreads();
        buf ^= 1;
    }

    // publish this block's partials
    float* P = ws + kWPART + blockIdx.x * kPSZ;
    for (int e = tid; e < kNK * kNF; e += 128) P[e] = csums[e];
    if (tid < kNK) P[kNK * kNF + tid] = ccnt[tid];
    if (tid == 0)  P[kNK * kNF + kNK] = potacc;
}

// ------------------------------------------- centers update + stop condition
__global__ void km_update_kernel(float* __restrict__ ws, int nblk) {
    if (ws[kWST] != 0.0f) return;
    __shared__ float cnts[kNK];
    __shared__ float newc[kNK * kNF];
    const int tid = threadIdx.x;

    if (tid < kNK) {
        float c = 0.0f;
        for (int b = 0; b < nblk; ++b) c += ws[kWPART + b * kPSZ + kNK * kNF + tid];
        cnts[tid] = c;
    }
    __syncthreads();
    for (int e = tid; e < kNK * kNF; e += 256) {
        float s = 0.0f;
        for (int b = 0; b < nblk; ++b) s += ws[kWPART + b * kPSZ + e];
        float v = s / cnts[e >> 6];        // 0/0 -> NaN for empty cluster (matches ref)
        newc[e] = v;
        ws[kWC + e] = v;
    }
    __syncthreads();
    if (tid < kNK) {
        float s = 0.0f;
        #pragma unroll 8
        for (int f = 0; f < kNF; ++f) { float v = newc[tid * kNF + f]; s += v * v; }
        ws[kWCN + tid] = s;
    }
    if (tid == 0) {
        float p = 0.0f;
        for (int b = 0; b < nblk; ++b) p += ws[kWPART + b * kPSZ + kNK * kNF + kNK];
        if (fabsf(p - ws[kWPP]) < kTol) ws[kWST] = 1.0f;
        ws[kWPP] = p;
    }
}

// --------------------------------------------------------------- final copy
__global__ void km_final_kernel(const float* __restrict__ ws, float* __restrict__ outc) {
    int g = blockIdx.x * blockDim.x + threadIdx.x;
    if (g < kNK * kNF) outc[g] = ws[kWC + g];
}

extern "C" void kernel_launch(void* const* d_in, const int* in_sizes, int n_in,
                              void* d_out, int out_size, void* d_ws, size_t ws_size,
                              hipStream_t stream) {
    const float* x     = (const float*)d_in[0];
    const float* minit = (const float*)d_in[1];
    float* ws = (float*)d_ws;
    int*   out_ixs = (int*)d_out;                     // int32 bit patterns, first kNS slots
    float* out_c   = (float*)d_out + kNS;             // centers after the indices

    // size the partial buffer to the workspace; prefer 1024 persistent blocks
    long long cap = ((long long)(ws_size / 4) - kWPART) / kPSZ;
    int nblk = (int)(cap < 1 ? 1 : (cap > 1024 ? 1024 : cap));

    km_init_kernel<<<16, 256, 0, stream>>>(minit, ws);
    km_xn_kernel<<<kNS / 8, 256, 0, stream>>>(x, ws);
    for (int it = 0; it < kIter; ++it) {
        km_assign_kernel<<<nblk, 128, 0, stream>>>(x, out_ixs, ws);
        km_update_kernel<<<1, 256, 0, stream>>>(ws, nblk);
    }
    km_final_kernel<<<16, 256, 0, stream>>>(ws, out_c);
}